// LambdaRankNdcgLoss_13692355739722
// MI455X (gfx1250) — compile-verified
//
#include <hip/hip_runtime.h>

#define NB 2048
#define NL 128

typedef __attribute__((ext_vector_type(2))) float v2f;
typedef __attribute__((ext_vector_type(8))) float v8f;

// One pairwise LambdaRank term from packed item data q = (pred, label, gain, disc):
//   s    = sign(l_i - l_j)
//   term = -|Δdisc * Δgain| * s * sigmoid(-s*(p_i - p_j))
// (s==0 => Δgain==0 => term is exactly 0, no branch needed)
// sigmoid via native v_exp_f32 + v_rcp_f32 (both ~1 ulp) instead of IEEE divide.
__device__ __forceinline__ float lr_term(const float4& qi, const float4& qj) {
    float s   = (float)((qi.y > qj.y) - (qi.y < qj.y));
    float pd  = s * (qi.x - qj.x);
    float lam = fabsf((qi.w - qj.w) * (qi.z - qj.z));
    float e   = __expf(pd);                       // native v_exp_f32
    float sig = __builtin_amdgcn_rcpf(1.0f + e);  // v_rcp_f32
    return -lam * s * sig;
}

__global__ __launch_bounds__(256) void lambdarank_ndcg_kernel(
    const float* __restrict__ logit, const float* __restrict__ label,
    float* __restrict__ out_loss, float* __restrict__ out_grad) {
    __shared__ float  sp[NL];      // pred (for rank loops)
    __shared__ float  sl[NL];      // label (for rank loops)
    __shared__ float  sld[NL];     // label position discount (idcg only)
    __shared__ float4 sq[NL];      // packed (pred, label, gain, pred_disc)
    __shared__ float  s_invidcg;

    const int b = blockIdx.x;
    const int t = threadIdx.x;     // 0..255, 8 wave32

    const float* __restrict__ pb = logit + (size_t)b * NL;
    const float* __restrict__ lb = label + (size_t)b * NL;

    if (t < NL) { sp[t] = pb[t]; sl[t] = lb[t]; }
    __syncthreads();

    // ---- Phase 1: ranks (stable descending sort semantics), discounts, gains ----
    if (t < NL) {
        const float pi = sp[t];
        int r = 0;
        #pragma unroll 8
        for (int j = 0; j < NL; ++j) {
            const float pj = sp[j];
            r += ((pj > pi) || ((pj == pi) && (j < t))) ? 1 : 0;
        }
        const float disc = 1.0f / log2f((float)(r + 2));
        const float gain = exp2f(sl[t]);
        sq[t] = make_float4(pi, sl[t], gain, disc);
    } else {
        const int i = t - NL;
        const float li = sl[i];
        int r = 0;
        #pragma unroll 8
        for (int j = 0; j < NL; ++j) {
            const float lj = sl[j];
            r += ((lj > li) || ((lj == li) && (j < i))) ? 1 : 0;
        }
        sld[i] = 1.0f / log2f((float)(r + 2));
    }
    __syncthreads();

    // ---- Phase 2: deterministic serial dcg / idcg reduction + loss ----
    if (t == 0) {
        float dcg = 0.0f, idcg = 0.0f;
        for (int j = 0; j < NL; ++j) {
            dcg  += sq[j].z * sq[j].w;
            idcg += sq[j].z * sld[j];
        }
        idcg = fmaxf(idcg, 1e-5f);
        s_invidcg   = 1.0f / idcg;
        out_loss[b] = (idcg - dcg) / idcg;
    }
    __syncthreads();
    const float inv_idcg = s_invidcg;

    // ---- Phase 3: pairwise grad, row-summed on the matrix pipe ----
    // Wave w owns rows [16w, 16w+16). A-matrix (16x4 f32) layout:
    //   lanes 0-15 : M = lane,    VGPR0 -> K=0, VGPR1 -> K=1
    //   lanes 16-31: M = lane-16, VGPR0 -> K=2, VGPR1 -> K=3
    // B = all-ones => D = rowsum(A) + C (layout of B irrelevant).
    const int wv   = t >> 5;
    const int lane = t & 31;
    const int m    = lane & 15;
    const int i    = (wv << 4) + m;
    const int jsel = (lane >> 4) << 1;   // 0 for lanes 0-15, 2 for lanes 16-31

    const float4 qi = sq[i];

    v8f acc = {};
    v2f bones; bones.x = 1.0f; bones.y = 1.0f;

    #pragma unroll 4
    for (int jb = 0; jb < NL; jb += 4) {
        const int j0 = jb + jsel;
        v2f a;
        a.x = lr_term(qi, sq[j0]);
        a.y = lr_term(qi, sq[j0 + 1]);
        // D = A * ones(4x16) + C  -> chained row-sum accumulation
        acc = __builtin_amdgcn_wmma_f32_16x16x4_f32(
            /*neg_a=*/false, a, /*neg_b=*/false, bones,
            /*c_mod=*/(short)0, acc, /*reuse_a=*/false, /*reuse_b=*/false);
    }

    // C/D layout: VGPR v holds M=v (lanes 0-15) and M=v+8 (lanes 16-31);
    // all N columns are identical row-sums, so lane 0 / lane 16 write them.
    float* __restrict__ gout = out_grad + (size_t)b * NL + (wv << 4);
    if (lane == 0) {
        #pragma unroll
        for (int v = 0; v < 8; ++v) gout[v] = acc[v] * inv_idcg;
    } else if (lane == 16) {
        #pragma unroll
        for (int v = 0; v < 8; ++v) gout[8 + v] = acc[v] * inv_idcg;
    }
}

extern "C" void kernel_launch(void* const* d_in, const int* in_sizes, int n_in,
                              void* d_out, int out_size, void* d_ws, size_t ws_size,
                              hipStream_t stream) {
    const float* logit = (const float*)d_in[0];
    const float* label = (const float*)d_in[1];
    float* out_loss = (float*)d_out;        // [NB]
    float* out_grad = out_loss + NB;        // [NB*NL]
    (void)in_sizes; (void)n_in; (void)out_size; (void)d_ws; (void)ws_size;
    hipLaunchKernelGGL(lambdarank_ndcg_kernel, dim3(NB), dim3(256), 0, stream,
                       logit, label, out_loss, out_grad);
}